// QuantumCBM_20203526160547
// MI455X (gfx1250) — compile-verified
//
#include <hip/hip_runtime.h>
#include <math.h>

// ---------------------------------------------------------------------------
// QuantumCBM for MI455X (gfx1250, wave32)
//  k1: qfeat = relu(x@W1+b1)@W2+b2        -> V_WMMA_F32_16X16X4_F32
//  k2: 14-qubit statevector sim, one batch row per workgroup, state in LDS
//  k3: head MLP (7->32->10), scalar VALU
// ---------------------------------------------------------------------------

typedef __attribute__((ext_vector_type(2))) float v2f;
typedef __attribute__((ext_vector_type(8))) float v8f;

#define NQ   14
#define QDIM 16384       // 2^14 amplitudes
#define BATCH 1024

__device__ __forceinline__ v8f wmma_f32_16x16x4(v2f a, v2f b, v8f c) {
  // (neg_a, A, neg_b, B, c_mod, C, reuse_a, reuse_b) — f32 variant: NEG must be 0
  return __builtin_amdgcn_wmma_f32_16x16x4_f32(false, a, false, b, (short)0, c,
                                               false, false);
}

// ---------------------------------------------------------------------------
// Kernel 1: qfeat (1024x14) = relu(x(1024x512) @ W1(512x64) + b1) @ W2(64x14) + b2
// grid = 64 row-blocks of 16; block = 128 threads (4 waves).
// Wave w computes the 16x16 hidden tile for columns [16w,16w+16) via WMMA,
// stages relu(hidden) into LDS, then wave 0 does the K=64 second GEMM.
// ---------------------------------------------------------------------------
__global__ __launch_bounds__(128) void qcbm_qfeat_kernel(
    const float* __restrict__ x, const float* __restrict__ W1,
    const float* __restrict__ b1, const float* __restrict__ W2,
    const float* __restrict__ b2, float* __restrict__ qfeat) {
  __shared__ float hid[16 * 64];

  const int lane = threadIdx.x & 31;
  const int wave = threadIdx.x >> 5;
  const int row0 = blockIdx.x * 16;
  const int n    = lane & 15;             // M index for A frag, N index for B/C frags
  const int koff = (lane >> 4) << 1;      // lanes 0-15: K=0,1 ; lanes 16-31: K=2,3
  const int mhi  = (lane >> 4) * 8;       // C frag: VGPR r -> row r (lo half) / r+8 (hi half)
  const int col0 = wave * 16;

  // ---- GEMM1: 16x16 tile, K=512 in steps of 4 ----
  v8f acc = {};
  const float* xrow = x + (size_t)(row0 + n) * 512;
  for (int k = 0; k < 512; k += 4) {
    v2f a, b;
    a[0] = xrow[k + koff];
    a[1] = xrow[k + koff + 1];
    b[0] = W1[(k + koff) * 64 + col0 + n];
    b[1] = W1[(k + koff + 1) * 64 + col0 + n];
    acc = wmma_f32_16x16x4(a, b, acc);
  }

  // bias + relu -> LDS (hidden tile is 16x64 across the 4 waves)
  const float bias = b1[col0 + n];
#pragma unroll
  for (int r = 0; r < 8; ++r) {
    float h = acc[r] + bias;
    hid[(r + mhi) * 64 + col0 + n] = h > 0.f ? h : 0.f;
  }
  __syncthreads();

  // ---- GEMM2: hidden(16x64) @ W2(64x14), padded to 16 cols; wave 0 only ----
  if (wave == 0) {
    v8f acc2 = {};
    for (int k = 0; k < 64; k += 4) {
      v2f a, b;
      a[0] = hid[n * 64 + k + koff];
      a[1] = hid[n * 64 + k + koff + 1];
      b[0] = (n < 14) ? W2[(k + koff) * 14 + n] : 0.f;
      b[1] = (n < 14) ? W2[(k + koff + 1) * 14 + n] : 0.f;
      acc2 = wmma_f32_16x16x4(a, b, acc2);
    }
    if (n < 14) {
      const float b2v = b2[n];
#pragma unroll
      for (int r = 0; r < 8; ++r)
        qfeat[(size_t)(row0 + r + mhi) * 14 + n] = acc2[r] + b2v;
    }
  }
}

// ---------------------------------------------------------------------------
// Kernel 2: quantum circuit + concept readout. One workgroup per batch row,
// full 128 KB statevector lives in LDS.
//
// Circuit algebra:
//  * RY(feat)+RY(qparams) layers on |0..0> -> product state, amp(i) = prod_q f_q(bit_q)
//  * CNOT block 1 (c=2c,t=2c+1, disjoint pairs)  -> index perm sigma
//  * RZ layer (diagonal)                         -> single phase per amplitude
//  * CNOT block 2 (c=2c+1,t=2c+2, disjoint)      -> index perm tau
//  => st[i] = exp(i*phase(tau(i))) * prod_q f_q(bit_q(sigma(tau(i))))   (no sweeps!)
//  * final RY layer -> 14 genuine LDS butterfly sweeps
//  * concepts[c] = sum_{bits 2c,2c+1 set} |amp|^2
// ---------------------------------------------------------------------------
__global__ __launch_bounds__(256) void qcbm_quantum_kernel(
    const float* __restrict__ qfeat, const float* __restrict__ qp,
    float* __restrict__ concepts_out) {
  __shared__ float2 st[QDIM];          // 128 KB statevector
  __shared__ float  sred[7][256];      // concept reduction scratch

  const int b   = blockIdx.x;
  const int tid = threadIdx.x;

  // per-thread gate tables (unrolled -> registers)
  float cs[NQ], sn[NQ], ph[NQ];
#pragma unroll
  for (int q = 0; q < NQ; ++q) {
    float th = 0.5f * (qfeat[b * NQ + q] + qp[q]);   // RY angles add on same qubit
    sincosf(th, &sn[q], &cs[q]);
    ph[q] = 0.5f * qp[NQ + q];                       // RZ half-angles
  }

  // ---- fused init: product state o sigma, phase, o tau ----
  for (int i = tid; i < QDIM; i += 256) {
    // tau: flip even target bit t in {2,4,..,12} if control bit t-1 of i is set
    int j = i ^ (((i >> 1) & 1) << 2) ^ (((i >> 3) & 1) << 4) ^ (((i >> 5) & 1) << 6)
              ^ (((i >> 7) & 1) << 8) ^ (((i >> 9) & 1) << 10) ^ (((i >> 11) & 1) << 12);
    // sigma: flip odd target bit c+1 if control bit c in {0,2,..,12} of j is set
    int m = j ^ (((j >> 0) & 1) << 1) ^ (((j >> 2) & 1) << 3) ^ (((j >> 4) & 1) << 5)
              ^ (((j >> 6) & 1) << 7) ^ (((j >> 8) & 1) << 9) ^ (((j >> 10) & 1) << 11)
              ^ (((j >> 12) & 1) << 13);
    float amp = 1.f, phase = 0.f;
#pragma unroll
    for (int q = 0; q < NQ; ++q) {
      amp   *= ((m >> q) & 1) ? sn[q] : cs[q];
      phase += ((j >> q) & 1) ? ph[q] : -ph[q];
    }
    float sp, cp;
    sincosf(phase, &sp, &cp);
    st[i] = make_float2(amp * cp, amp * sp);
  }

  // ---- final RY layer: 14 butterfly sweeps over LDS ----
  for (int q = 0; q < NQ; ++q) {
    __syncthreads();
    float hc, hs;
    sincosf(0.5f * qp[2 * NQ + q], &hs, &hc);
    const int mask = (1 << q) - 1;
    for (int p = tid; p < QDIM / 2; p += 256) {
      int i0 = ((p & ~mask) << 1) | (p & mask);
      int i1 = i0 | (1 << q);
      float2 a0 = st[i0], a1 = st[i1];
      st[i0] = make_float2(hc * a0.x - hs * a1.x, hc * a0.y - hs * a1.y);
      st[i1] = make_float2(hs * a0.x + hc * a1.x, hs * a0.y + hc * a1.y);
    }
  }
  __syncthreads();

  // ---- concept readout: concepts[c] = sum over {bit 2c & bit 2c+1} |amp|^2 ----
  float acc[7] = {0.f, 0.f, 0.f, 0.f, 0.f, 0.f, 0.f};
  for (int i = tid; i < QDIM; i += 256) {
    float2 v = st[i];
    float p = v.x * v.x + v.y * v.y;
#pragma unroll
    for (int c = 0; c < 7; ++c)
      if (((i >> (2 * c)) & 3) == 3) acc[c] += p;
  }
#pragma unroll
  for (int c = 0; c < 7; ++c) sred[c][tid] = acc[c];
  for (int off = 128; off >= 1; off >>= 1) {
    __syncthreads();
    if (tid < off) {
#pragma unroll
      for (int c = 0; c < 7; ++c) sred[c][tid] += sred[c][tid + off];
    }
  }
  __syncthreads();
  if (tid < 7) concepts_out[b * 7 + tid] = sred[tid][0];
}

// ---------------------------------------------------------------------------
// Kernel 3: logits = relu(concepts@Wp1+bp1)@Wp2+bp2  (one thread per row)
// ---------------------------------------------------------------------------
__global__ __launch_bounds__(256) void qcbm_head_kernel(
    const float* __restrict__ concepts, const float* __restrict__ Wp1,
    const float* __restrict__ bp1, const float* __restrict__ Wp2,
    const float* __restrict__ bp2, float* __restrict__ logits) {
  const int r = blockIdx.x * 256 + threadIdx.x;
  if (r >= BATCH) return;
  float c[7];
#pragma unroll
  for (int i = 0; i < 7; ++i) c[i] = concepts[r * 7 + i];
  float h[32];
#pragma unroll
  for (int j = 0; j < 32; ++j) {
    float s = bp1[j];
#pragma unroll
    for (int i = 0; i < 7; ++i) s += c[i] * Wp1[i * 32 + j];
    h[j] = s > 0.f ? s : 0.f;
  }
#pragma unroll
  for (int k = 0; k < 10; ++k) {
    float s = bp2[k];
#pragma unroll
    for (int j = 0; j < 32; ++j) s += h[j] * Wp2[j * 10 + k];
    logits[r * 10 + k] = s;
  }
}

// ---------------------------------------------------------------------------
extern "C" void kernel_launch(void* const* d_in, const int* in_sizes, int n_in,
                              void* d_out, int out_size, void* d_ws, size_t ws_size,
                              hipStream_t stream) {
  const float* x   = (const float*)d_in[0];
  const float* W1  = (const float*)d_in[1];
  const float* b1  = (const float*)d_in[2];
  const float* W2  = (const float*)d_in[3];
  const float* b2  = (const float*)d_in[4];
  const float* qp  = (const float*)d_in[5];
  const float* Wp1 = (const float*)d_in[6];
  const float* bp1 = (const float*)d_in[7];
  const float* Wp2 = (const float*)d_in[8];
  const float* bp2 = (const float*)d_in[9];

  float* out      = (float*)d_out;
  float* logits   = out;                  // 1024*10 floats
  float* concepts = out + BATCH * 10;     // 1024*7 floats
  float* qfeat    = (float*)d_ws;         // 1024*14 floats scratch

  qcbm_qfeat_kernel<<<dim3(64), dim3(128), 0, stream>>>(x, W1, b1, W2, b2, qfeat);
  qcbm_quantum_kernel<<<dim3(BATCH), dim3(256), 0, stream>>>(qfeat, qp, concepts);
  qcbm_head_kernel<<<dim3(4), dim3(256), 0, stream>>>(concepts, Wp1, bp1, Wp2, bp2,
                                                      logits);
}